// SSRM_35338990911668
// MI455X (gfx1250) — compile-verified
//
#include <hip/hip_runtime.h>

// ---------------------------------------------------------------------------
// Sorted-window attention module for MI455X (gfx1250), fp32 WMMA 16x16x4.
// N=2, Cin=200, H=W=128, L=16384, scales=4, window=64, C=50 (pad 52), 4C=200.
// ---------------------------------------------------------------------------

typedef float v2f __attribute__((ext_vector_type(2)));
typedef float v8f __attribute__((ext_vector_type(8)));

#define LANE (threadIdx.x & 31)
#define WAVE (threadIdx.x >> 5)

static __device__ __forceinline__ v8f zero8() {
  v8f z;
#pragma unroll
  for (int i = 0; i < 8; ++i) z[i] = 0.0f;
  return z;
}

static __device__ __forceinline__ v8f wmma4(v2f a, v2f b, v8f c) {
#if defined(__gfx1250__)
  // D(16x16,f32) = A(16x4,f32) * B(4x16,f32) + C   -> v_wmma_f32_16x16x4_f32
  return __builtin_amdgcn_wmma_f32_16x16x4_f32(false, a, false, b, (short)0, c,
                                               false, false);
#else
  c[0] += a.x * b.x;  // host-pass stub, never executed on device
  return c;
#endif
}

// Async gathered copy global -> LDS (16B), tracked by ASYNCcnt. Per-lane
// addresses; no VGPR staging of the data itself.
static __device__ __forceinline__ void async_g2l_b128(unsigned int ldsaddr,
                                                      const void* gaddr) {
#if defined(__gfx1250__)
  asm volatile("global_load_async_to_lds_b128 %0, %1, off"
               :
               : "v"(ldsaddr), "v"(gaddr)
               : "memory");
#else
  (void)ldsaddr;
  (void)gaddr;
#endif
}

static __device__ __forceinline__ void wait_async0() {
#if defined(__gfx1250__)
  asm volatile("s_wait_asynccnt 0x0" ::: "memory");
#endif
}

static __device__ __forceinline__ int refl(int i) {  // 'reflect' pad for [0,128)
  return i < 0 ? -i : (i > 127 ? 254 - i : i);
}

// ---------------------------------------------------------------------------
// Weight repack: conv weights -> [k=(tap*200+ci)][o pad 64]
//                Wy -> [ci][o pad 208], Wout -> [q][o pad 208]
// ---------------------------------------------------------------------------
__global__ void k_prep(const float* __restrict__ Wx, const float* __restrict__ Wy,
                       const float* __restrict__ Wout, float* __restrict__ WgC,
                       float* __restrict__ WgY, float* __restrict__ WgO) {
  int i = blockIdx.x * 256 + threadIdx.x;
  if (i < 1800 * 64) {
    int kk = i >> 6, o = i & 63;
    int tap = kk / 200, ci = kk % 200;
    WgC[i] = (o < 50) ? Wx[(o * 200 + ci) * 9 + tap] : 0.0f;
  }
  if (i < 200 * 208) {
    int ci = i / 208, o = i % 208;
    WgY[i] = (o < 200) ? Wy[o * 200 + ci] : 0.0f;
  }
  if (i < 800 * 208) {
    int q = i / 208, o = i % 208;
    WgO[i] = (o < 200) ? Wout[o * 800 + q] : 0.0f;
  }
}

// ---------------------------------------------------------------------------
// 3x3 reflect conv as implicit GEMM: M=16 pixels/wave, N=64 (50 real), K=1800.
// x_embed layout: [n*16384 + l][52]  (cols 50,51 are zeros)
// ---------------------------------------------------------------------------
__global__ void k_conv(const float* __restrict__ in, const float* __restrict__ Wg,
                       float* __restrict__ xe) {
  int gw = blockIdx.x * 4 + WAVE;  // 2048 waves
  int n = gw >> 10;
  int l0 = (gw & 1023) << 4;
  int h = l0 >> 7, w0 = l0 & 127;
  int lane = LANE, mcol = lane & 15, ksub = (lane >> 4) << 1;
  v8f a0 = zero8(), a1 = zero8(), a2 = zero8(), a3 = zero8();
  const float* inb = in + (size_t)n * 200 * 16384;
  for (int tap = 0; tap < 9; ++tap) {
    int dy = tap / 3, dx = tap % 3;
    int hs = refl(h + dy - 1);
    int wsrc = refl(w0 + mcol + dx - 1);
    int pix = hs * 128 + wsrc;
    int kt0 = tap * 200;
    for (int cs = 0; cs < 50; ++cs) {
      int ci = cs * 4 + ksub;
      v2f a;
      a.x = inb[(size_t)ci * 16384 + pix];
      a.y = inb[(size_t)(ci + 1) * 16384 + pix];
      const float* wp = Wg + (size_t)(kt0 + cs * 4 + ksub) * 64 + mcol;
      v2f b0, b1, b2, b3;
      b0.x = wp[0];  b0.y = wp[64];
      b1.x = wp[16]; b1.y = wp[80];
      b2.x = wp[32]; b2.y = wp[96];
      b3.x = wp[48]; b3.y = wp[112];
      a0 = wmma4(a, b0, a0); a1 = wmma4(a, b1, a1);
      a2 = wmma4(a, b2, a2); a3 = wmma4(a, b3, a3);
    }
  }
  int rbase = (lane >> 4) * 8;
  float* xrow = xe + ((size_t)n * 16384 + l0) * 52;
#pragma unroll
  for (int v = 0; v < 8; ++v) {
    float* r = xrow + (size_t)(rbase + v) * 52;
    r[mcol] = a0[v];
    r[16 + mcol] = a1[v];
    r[32 + mcol] = a2[v];
    if (mcol < 4) r[48 + mcol] = a3[v];  // cols 48..51 only (50,51 = zero pad)
  }
}

// ---------------------------------------------------------------------------
// y_embed = input^T * Wy : M=16/wave, N=208 (200 real), K=200.
// y_embed layout: [n*16384 + l][200]
// ---------------------------------------------------------------------------
__global__ void k_ygemm(const float* __restrict__ in, const float* __restrict__ Wg,
                        float* __restrict__ ye) {
  int gw = blockIdx.x * 4 + WAVE;
  int n = gw >> 10;
  int l0 = (gw & 1023) << 4;
  int lane = LANE, mcol = lane & 15, ksub = (lane >> 4) << 1;
  v8f acc[13];
#pragma unroll
  for (int i = 0; i < 13; ++i) acc[i] = zero8();
  const float* inb = in + (size_t)n * 200 * 16384 + l0 + mcol;
  for (int cs = 0; cs < 50; ++cs) {
    int ci = cs * 4 + ksub;
    v2f a;
    a.x = inb[(size_t)ci * 16384];
    a.y = inb[(size_t)(ci + 1) * 16384];
    const float* wp = Wg + (size_t)ci * 208 + mcol;
#pragma unroll
    for (int nt = 0; nt < 13; ++nt) {
      v2f b;
      b.x = wp[nt * 16];
      b.y = wp[208 + nt * 16];
      acc[nt] = wmma4(a, b, acc[nt]);
    }
  }
  int rbase = (lane >> 4) * 8;
  float* yrow = ye + ((size_t)n * 16384 + l0) * 200;
#pragma unroll
  for (int nt = 0; nt < 13; ++nt) {
    int col = nt * 16 + mcol;
    if (col < 200) {
#pragma unroll
      for (int v = 0; v < 8; ++v) yrow[(size_t)(rbase + v) * 200 + col] = acc[nt][v];
    }
  }
}

// 1 / max(||x_row||_2, 5e-5) per pixel
__global__ void k_rnorm(const float* __restrict__ xe, float* __restrict__ rinv) {
  int r = blockIdx.x * 256 + threadIdx.x;
  if (r >= 32768) return;
  const float* p = xe + (size_t)r * 52;
  float s = 0.0f;
  for (int c = 0; c < 50; ++c) { float v = p[c]; s += v * v; }
  rinv[r] = 1.0f / fmaxf(sqrtf(s), 5e-5f);
}

// ---------------------------------------------------------------------------
// Stable counting sort over 128 buckets (scale*32 + label) == jnp.argsort(sl)
// ---------------------------------------------------------------------------
__global__ void k_hist(const int* __restrict__ lab, int* __restrict__ counts) {
  __shared__ int cnt[128];
  int n = blockIdx.x >> 7, blk = blockIdx.x & 127;
  if (threadIdx.x < 128) cnt[threadIdx.x] = 0;
  __syncthreads();
  int j = blk * 512 + (int)threadIdx.x;
  int s = j >> 14;
  int b = (s << 5) + lab[n * 65536 + j];
  atomicAdd(&cnt[b], 1);
  __syncthreads();
  if (threadIdx.x < 128)
    counts[n * 16384 + (int)threadIdx.x * 128 + blk] = cnt[threadIdx.x];
}

__global__ void k_scan(const int* __restrict__ counts, int* __restrict__ offs) {
  __shared__ int part[512];
  int n = blockIdx.x, t = threadIdx.x;
  const int* c = counts + n * 16384;
  int* o = offs + n * 16384;
  int base = t * 32, s = 0;
  for (int i = 0; i < 32; ++i) s += c[base + i];
  part[t] = s;
  __syncthreads();
  for (int d = 1; d < 512; d <<= 1) {
    int tmp = (t >= d) ? part[t - d] : 0;
    __syncthreads();
    part[t] += tmp;
    __syncthreads();
  }
  int run = part[t] - s;  // exclusive prefix
  for (int i = 0; i < 32; ++i) { o[base + i] = run; run += c[base + i]; }
}

__global__ void k_scatter(const int* __restrict__ lab, const int* __restrict__ offs,
                          int* __restrict__ pos, int* __restrict__ inv) {
  __shared__ unsigned char kb[512];
  int n = blockIdx.x >> 7, blk = blockIdx.x & 127;
  int t = threadIdx.x;
  int j = blk * 512 + t;
  int s = j >> 14;
  int b = (s << 5) + lab[n * 65536 + j];
  kb[t] = (unsigned char)b;
  __syncthreads();
  int rank = 0;
  for (int e = 0; e < t; ++e) rank += (kb[e] == (unsigned char)b);
  int p = offs[n * 16384 + b * 128 + blk] + rank;
  pos[n * 65536 + j] = p;       // per-batch sorted position of element j
  inv[n * 65536 + p] = j;       // inverse permutation
}

// ---------------------------------------------------------------------------
// Fused attention: one 256-thread workgroup per cluster (2048 clusters).
// S = Xq(64x52) * (Xk*rinv)^T(52x192) ; softmax ; out = P(64x192) * Y(192x200)
// x staging uses gfx1250 async global->LDS DMA (no VGPR round-trip).
// ---------------------------------------------------------------------------
__global__ void __launch_bounds__(256) k_attn(
    const float* __restrict__ xe, const float* __restrict__ ye,
    const float* __restrict__ rinv, const int* __restrict__ inv,
    float* __restrict__ attn) {
  __shared__ float region[13312];  // phase1: xq(64x52)+xk(192x52); phase2: S(64x192)
  __shared__ int rowoff[192];      // per-key-row pixel row index (n*16384 + l)
  __shared__ float rowscale[192];  // per-key-row 1/||x||
  float* xq = region;
  float* xk = region + 3328;
  float* S = region;

  int cid = blockIdx.x;
  int n = cid >> 10;
  int s = (cid >> 8) & 3;
  int k = cid & 255;
  int segbase = n * 65536 + s * 16384;
  int p0 = segbase + k * 64;
  int t = threadIdx.x;

  if (t < 192) {
    int w = t >> 6;
    int kw = (w == 0) ? k : (w == 1) ? ((k + 255) & 255) : ((k + 1) & 255);
    int p = segbase + kw * 64 + (t & 63);
    int l = inv[p] & 16383;
    rowoff[t] = n * 16384 + l;
    rowscale[t] = rinv[n * 16384 + l];
    // warm the cache for the PV-phase gathered value rows
    __builtin_prefetch(ye + (size_t)(n * 16384 + l) * 200, 0, 0);
  }
  __syncthreads();

  // queries: raw rows, async DMA 16B chunks (row stride 208B -> 16B aligned)
  for (int q4 = t; q4 < 64 * 13; q4 += 256) {
    int r = q4 / 13, c4 = (q4 - r * 13) * 4;
    int l = inv[p0 + r] & 16383;
    async_g2l_b128((unsigned int)(size_t)&xq[r * 52 + c4],
                   xe + (size_t)(n * 16384 + l) * 52 + c4);
  }
  // keys: raw rows (normalization folded into B-fragment build below)
  for (int q4 = t; q4 < 192 * 13; q4 += 256) {
    int r = q4 / 13, c4 = (q4 - r * 13) * 4;
    async_g2l_b128((unsigned int)(size_t)&xk[r * 52 + c4],
                   xe + (size_t)rowoff[r] * 52 + c4);
  }
  wait_async0();
  __syncthreads();

  int lane = LANE, wv = WAVE;
  int mt = wv & 3, g = wv >> 2;
  int mcol = lane & 15, ksub = (lane >> 4) << 1, rbase = (lane >> 4) * 8;

  // ---- QK: wave (mt,g) computes M-tile mt, N-tiles g*6 .. g*6+5
  v8f sacc[6];
  float sc[6];
#pragma unroll
  for (int i = 0; i < 6; ++i) {
    sacc[i] = zero8();
    sc[i] = rowscale[(g * 6 + i) * 16 + mcol];  // per-key-row 1/||x||
  }
  for (int kt = 0; kt < 13; ++kt) {
    int ka = kt * 4 + ksub;
    int m = mt * 16 + mcol;
    v2f a;
    a.x = xq[m * 52 + ka];
    a.y = xq[m * 52 + ka + 1];
#pragma unroll
    for (int jn = 0; jn < 6; ++jn) {
      int nc = (g * 6 + jn) * 16 + mcol;  // key row
      v2f b;
      b.x = xk[nc * 52 + ka] * sc[jn];
      b.y = xk[nc * 52 + ka + 1] * sc[jn];
      sacc[jn] = wmma4(a, b, sacc[jn]);
    }
  }
  __syncthreads();  // all xq/xk reads complete before S overwrites the region
#pragma unroll
  for (int jn = 0; jn < 6; ++jn) {
    int col = (g * 6 + jn) * 16 + mcol;
#pragma unroll
    for (int v = 0; v < 8; ++v) S[(mt * 16 + rbase + v) * 192 + col] = sacc[jn][v];
  }
  __syncthreads();

  // ---- softmax over 192 per row
  if (t < 64) {
    float* row = S + t * 192;
    float mx = -1e30f;
    for (int i = 0; i < 192; ++i) mx = fmaxf(mx, row[i]);
    float sum = 0.0f;
    for (int i = 0; i < 192; ++i) { float e = __expf(row[i] - mx); row[i] = e; sum += e; }
    float is = 1.0f / sum;
    for (int i = 0; i < 192; ++i) row[i] *= is;
  }
  __syncthreads();

  // ---- PV: wave (mt,g): N-tiles g, g+2, ... ; B gathered from y_embed (L2-resident)
  v8f pacc[7];
#pragma unroll
  for (int i = 0; i < 7; ++i) pacc[i] = zero8();
  for (int kt = 0; kt < 48; ++kt) {
    int ka = kt * 4 + ksub;
    int m = mt * 16 + mcol;
    v2f a;
    a.x = S[m * 192 + ka];
    a.y = S[m * 192 + ka + 1];
    const float* y0 = ye + (size_t)rowoff[ka] * 200;
    const float* y1 = ye + (size_t)rowoff[ka + 1] * 200;
#pragma unroll
    for (int jn = 0; jn < 7; ++jn) {
      int col = (g + 2 * jn) * 16 + mcol;
      v2f b;
      b.x = (col < 200) ? y0[col] : 0.0f;
      b.y = (col < 200) ? y1[col] : 0.0f;
      pacc[jn] = wmma4(a, b, pacc[jn]);
    }
  }
#pragma unroll
  for (int jn = 0; jn < 7; ++jn) {
    int col = (g + 2 * jn) * 16 + mcol;
    if (col < 200) {
#pragma unroll
      for (int v = 0; v < 8; ++v)
        attn[(size_t)(p0 + mt * 16 + rbase + v) * 200 + col] = pacc[jn][v];
    }
  }
}

// ---------------------------------------------------------------------------
// Unsort + 1x1 conv (K=800) + bias + residual. M=16 pixels/wave, N=208.
// ---------------------------------------------------------------------------
__global__ void k_final(const float* __restrict__ attn, const float* __restrict__ Wg,
                        const int* __restrict__ pos, const float* __restrict__ bout,
                        const float* __restrict__ in, float* __restrict__ out) {
  int gw = blockIdx.x * 4 + WAVE;
  int n = gw >> 10;
  int l0 = (gw & 1023) << 4;
  int lane = LANE, mcol = lane & 15, ksub = (lane >> 4) << 1, rbase = (lane >> 4) * 8;
  int l = l0 + mcol;
  size_t rp[4];
#pragma unroll
  for (int s = 0; s < 4; ++s)
    rp[s] = ((size_t)n * 65536 + (size_t)pos[n * 65536 + s * 16384 + l]) * 200;
  v8f acc[13];
#pragma unroll
  for (int i = 0; i < 13; ++i) acc[i] = zero8();
  for (int s = 0; s < 4; ++s) {
    const float* arow = attn + rp[s];
    for (int cs = 0; cs < 50; ++cs) {
      int ka = cs * 4 + ksub;
      v2f a;
      a.x = arow[ka];
      a.y = arow[ka + 1];
      const float* wp = Wg + (size_t)(s * 200 + ka) * 208 + mcol;
#pragma unroll
      for (int nt = 0; nt < 13; ++nt) {
        v2f b;
        b.x = wp[nt * 16];
        b.y = wp[208 + nt * 16];
        acc[nt] = wmma4(a, b, acc[nt]);
      }
    }
  }
#pragma unroll
  for (int nt = 0; nt < 13; ++nt) {
    int o = nt * 16 + mcol;
    if (o < 200) {
      float bo = bout[o];
#pragma unroll
      for (int v = 0; v < 8; ++v) {
        size_t idx = ((size_t)n * 200 + o) * 16384 + l0 + rbase + v;
        out[idx] = acc[nt][v] + bo + in[idx];
      }
    }
  }
}

// ---------------------------------------------------------------------------
extern "C" void kernel_launch(void* const* d_in, const int* in_sizes, int n_in,
                              void* d_out, int out_size, void* d_ws, size_t ws_size,
                              hipStream_t stream) {
  const float* input = (const float*)d_in[0];
  const int* lab = (const int*)d_in[1];
  const float* Wx = (const float*)d_in[2];
  const float* Wy = (const float*)d_in[3];
  const float* Wout = (const float*)d_in[4];
  const float* bout = (const float*)d_in[5];
  float* out = (float*)d_out;

  char* ws = (char*)d_ws;
  size_t off = 0;
  auto alloc = [&](size_t bytes) -> char* {
    char* p = ws + off;
    off = (off + bytes + 255) & ~(size_t)255;
    return p;
  };
  float* WgC = (float*)alloc((size_t)1800 * 64 * 4);     // conv weights repacked
  float* WgY = (float*)alloc((size_t)200 * 208 * 4);     // 1x1 y weights
  float* WgO = (float*)alloc((size_t)800 * 208 * 4);     // output 1x1 weights
  float* xe = (float*)alloc((size_t)32768 * 52 * 4);     // x_embed [NL][52]
  float* ye = (float*)alloc((size_t)32768 * 200 * 4);    // y_embed [NL][200]
  float* rinv = (float*)alloc((size_t)32768 * 4);        // 1/||x|| per pixel
  int* counts = (int*)alloc((size_t)32768 * 4);          // [n][bucket*128+block]
  int* offs = (int*)alloc((size_t)32768 * 4);            // scanned
  int* pos = (int*)alloc((size_t)131072 * 4);            // forward permutation
  int* inv = (int*)alloc((size_t)131072 * 4);            // inverse permutation
  float* attn = (float*)alloc((size_t)131072 * 200 * 4); // sorted attention out
  (void)ws_size; (void)in_sizes; (void)n_in; (void)out_size;

  k_prep<<<650, 256, 0, stream>>>(Wx, Wy, Wout, WgC, WgY, WgO);
  k_conv<<<512, 128, 0, stream>>>(input, WgC, xe);
  k_ygemm<<<512, 128, 0, stream>>>(input, WgY, ye);
  k_rnorm<<<128, 256, 0, stream>>>(xe, rinv);
  k_hist<<<256, 512, 0, stream>>>(lab, counts);
  k_scan<<<2, 512, 0, stream>>>(counts, offs);
  k_scatter<<<256, 512, 0, stream>>>(lab, offs, pos, inv);
  k_attn<<<2048, 256, 0, stream>>>(xe, ye, rinv, inv, attn);
  k_final<<<512, 128, 0, stream>>>(attn, WgO, pos, bout, input, out);
}